// GAT_66511863546091
// MI455X (gfx1250) — compile-verified
//
#include <hip/hip_runtime.h>

#define N_NODES   100000
#define N_EDGES   6400000
#define F_IN      128
#define NEG_SLOPE 0.2f
#define N_TILES   (N_NODES / 16)   // 6250 exactly, no remainder

typedef __attribute__((ext_vector_type(2))) float v2f;
typedef __attribute__((ext_vector_type(8))) float v8f;

// ---------------------------------------------------------------------------
// Float atomic max via sign-split ordered-int trick (lowers to native
// GLOBAL_ATOMIC_MAX_I32 / GLOBAL_ATOMIC_MIN_U32 — no CAS loop).
// Requires *addr initialized to -inf.
// ---------------------------------------------------------------------------
__device__ __forceinline__ void atomicMaxFloat(float* addr, float val) {
    if (val >= 0.0f) {
        atomicMax((int*)addr, __float_as_int(val));
    } else {
        atomicMin((unsigned int*)addr, __float_as_uint(val));
    }
}

// ---------------------------------------------------------------------------
// Kernel 0: init amax=-inf, denom=0, out=bias
// ---------------------------------------------------------------------------
__global__ void gat_init(float* __restrict__ amax, float* __restrict__ denom,
                         float* __restrict__ out, const float* __restrict__ bias) {
    int i = blockIdx.x * blockDim.x + threadIdx.x;
    if (i < N_NODES) {
        amax[i]  = -__builtin_huge_valf();
        denom[i] = 0.0f;
        out[(size_t)i * 2 + 0] = bias[0];
        out[(size_t)i * 2 + 1] = bias[1];
    }
}

// ---------------------------------------------------------------------------
// Kernel 1: fused node GEMM via V_WMMA_F32_16X16X4_F32.
// node_data[n] = (h0, h1, a_src, a_dst) = x[n,:] @ Wc, where
// Wc[:,0:2] = W_src, Wc[:,2] = W_src@att_src, Wc[:,3] = W_src@att_dst.
// One 16-row node tile per wave; K=128 -> 32 accumulating WMMAs.
//
// A (16x4 f32) layout: lanes 0-15 row M=lane, v0/v1 = K,K+1;
//                      lanes 16-31 row M=lane-16, v0/v1 = K+2,K+3.
// B (4x16 f32) layout: v0 lanes 0-15 = row K (N=lane), lanes 16-31 = row K+2;
//                      v1 = rows K+1 / K+3.
// D (16x16 f32): vgpr r, lane L -> row r + (L>=16 ? 8 : 0), col L&15.
// ---------------------------------------------------------------------------
__global__ void gat_node_wmma(const float* __restrict__ x,
                              const float* __restrict__ W_src,    // [128][2]
                              const float* __restrict__ att_src,  // [2]
                              const float* __restrict__ att_dst,  // [2]
                              float* __restrict__ node_data) {    // [N][4]
    __shared__ float Wc[F_IN][16];   // 8 KB
    const int t = threadIdx.x;       // 256 threads = 8 waves

    // Build fused weight matrix in LDS (cols 4..15 are zero padding).
    for (int i = t; i < F_IN * 16; i += 256) {
        int k = i >> 4, n = i & 15;
        float w0 = W_src[k * 2 + 0], w1 = W_src[k * 2 + 1];
        float v = 0.0f;
        if      (n == 0) v = w0;
        else if (n == 1) v = w1;
        else if (n == 2) v = w0 * att_src[0] + w1 * att_src[1];
        else if (n == 3) v = w0 * att_dst[0] + w1 * att_dst[1];
        Wc[k][n] = v;
    }
    __syncthreads();

    const int lane = t & 31;
    const int wave = t >> 5;
    const int tile = blockIdx.x * 8 + wave;   // wave-uniform
    if (tile >= N_TILES) return;              // uniform exit: EXEC stays full

    const int n    = lane & 15;               // column within tile / B column
    const int koff = (lane >> 4) * 2;         // half-wave K split (0 or 2)
    const int m    = tile * 16 + n;           // node row this lane feeds for A
    const float* xrow = x + (size_t)m * F_IN;

    v8f acc = {};
    #pragma unroll 8
    for (int k = 0; k < F_IN; k += 4) {
        v2f a, b;
        a.x = xrow[k + koff];
        a.y = xrow[k + koff + 1];
        b.x = Wc[k + koff][n];
        b.y = Wc[k + koff + 1][n];
        acc = __builtin_amdgcn_wmma_f32_16x16x4_f32(
            /*neg_a=*/false, a, /*neg_b=*/false, b,
            /*c_mod=*/(short)0, acc, /*reuse_a=*/false, /*reuse_b=*/false);
    }

    // Lanes whose column < 4 own valid node_data columns; each holds 8 rows.
    if (n < 4) {
        const int rbase = tile * 16 + (lane >> 4) * 8;
        #pragma unroll
        for (int r = 0; r < 8; ++r)
            node_data[(size_t)(rbase + r) * 4 + n] = acc[r];
    }
}

// ---------------------------------------------------------------------------
// Kernel 2: per-edge logits + leaky_relu, atomic segment-max over dst.
// ---------------------------------------------------------------------------
__global__ void gat_edge_logits(const int* __restrict__ src,
                                const int* __restrict__ dst,
                                const float* __restrict__ edge_attr,
                                const float* __restrict__ W_edge,   // [2]
                                const float* __restrict__ att_edge, // [2]
                                const float* __restrict__ node_data,
                                float* __restrict__ alpha_ws,
                                float* __restrict__ amax) {
    const float c_e = W_edge[0] * att_edge[0] + W_edge[1] * att_edge[1];
    const int stride = gridDim.x * blockDim.x;
    for (int e = blockIdx.x * blockDim.x + threadIdx.x; e < N_EDGES; e += stride) {
        int s = src[e], d = dst[e];
        float al = node_data[(size_t)s * 4 + 2]     // a_src[s]
                 + node_data[(size_t)d * 4 + 3]     // a_dst[d]
                 + c_e * edge_attr[e];              // a_e
        al = al > 0.0f ? al : NEG_SLOPE * al;
        alpha_ws[e] = al;
        atomicMaxFloat(&amax[d], al);
    }
}

// ---------------------------------------------------------------------------
// Kernel 3: exponentiate shifted logits, atomic segment-sum (denominator).
// ---------------------------------------------------------------------------
__global__ void gat_edge_exp(const int* __restrict__ dst,
                             const float* __restrict__ amax,
                             float* __restrict__ alpha_ws,
                             float* __restrict__ denom) {
    const int stride = gridDim.x * blockDim.x;
    for (int e = blockIdx.x * blockDim.x + threadIdx.x; e < N_EDGES; e += stride) {
        int d = dst[e];
        float ex = __expf(alpha_ws[e] - amax[d]);
        alpha_ws[e] = ex;
        atomicAdd(&denom[d], ex);
    }
}

// ---------------------------------------------------------------------------
// Kernel 4: normalize, emit alpha, weighted scatter-add of h[src] into out.
// ---------------------------------------------------------------------------
__global__ void gat_edge_agg(const int* __restrict__ src,
                             const int* __restrict__ dst,
                             const float* __restrict__ alpha_ws,
                             const float* __restrict__ denom,
                             const float* __restrict__ node_data,
                             float* __restrict__ out,        // [N][2]
                             float* __restrict__ alpha_out)  // [E]
{
    const int stride = gridDim.x * blockDim.x;
    for (int e = blockIdx.x * blockDim.x + threadIdx.x; e < N_EDGES; e += stride) {
        int s = src[e], d = dst[e];
        float an = alpha_ws[e] / (denom[d] + 1e-16f);
        alpha_out[e] = an;
        atomicAdd(&out[(size_t)d * 2 + 0], node_data[(size_t)s * 4 + 0] * an);
        atomicAdd(&out[(size_t)d * 2 + 1], node_data[(size_t)s * 4 + 1] * an);
    }
}

// ---------------------------------------------------------------------------
// Host-side launcher. Inputs (setup_inputs order):
//  0:x [N,128] f32   1:edge_index [2,E] int   2:edge_attr [E,1] f32
//  3:W_src [128,2]   4:W_edge [1,2]   5:att_src [2]   6:att_dst [2]
//  7:att_edge [2]    8:bias [2]
// d_out = [ out (N*2 f32) | alpha (E f32) ]
// ---------------------------------------------------------------------------
extern "C" void kernel_launch(void* const* d_in, const int* in_sizes, int n_in,
                              void* d_out, int out_size, void* d_ws, size_t ws_size,
                              hipStream_t stream) {
    const float* x         = (const float*)d_in[0];
    const int*   edge_idx  = (const int*)d_in[1];
    const float* edge_attr = (const float*)d_in[2];
    const float* W_src     = (const float*)d_in[3];
    const float* W_edge    = (const float*)d_in[4];
    const float* att_src   = (const float*)d_in[5];
    const float* att_dst   = (const float*)d_in[6];
    const float* att_edge  = (const float*)d_in[7];
    const float* bias      = (const float*)d_in[8];

    const int* src = edge_idx;            // row 0
    const int* dst = edge_idx + N_EDGES;  // row 1

    float* out       = (float*)d_out;                 // [N,2]
    float* alpha_out = out + (size_t)N_NODES * 2;     // [E]

    // Workspace layout (28 MB total)
    float* node_data = (float*)d_ws;                       // N*4
    float* amax      = node_data + (size_t)N_NODES * 4;    // N
    float* denom     = amax + N_NODES;                     // N
    float* alpha_ws  = denom + N_NODES;                    // E
    (void)ws_size; (void)n_in; (void)in_sizes; (void)out_size;

    const int TPB = 256;

    // 0) init
    gat_init<<<(N_NODES + TPB - 1) / TPB, TPB, 0, stream>>>(amax, denom, out, bias);

    // 1) fused node GEMM (WMMA): node_data = x @ [W | W@att_src | W@att_dst]
    gat_node_wmma<<<(N_TILES + 7) / 8, TPB, 0, stream>>>(x, W_src, att_src, att_dst,
                                                         node_data);

    // 2) logits + segment max
    const int eblocks = (N_EDGES + TPB - 1) / TPB;
    gat_edge_logits<<<eblocks, TPB, 0, stream>>>(src, dst, edge_attr, W_edge,
                                                 att_edge, node_data, alpha_ws, amax);

    // 3) exp + segment sum
    gat_edge_exp<<<eblocks, TPB, 0, stream>>>(dst, amax, alpha_ws, denom);

    // 4) normalize + scatter-add aggregate
    gat_edge_agg<<<eblocks, TPB, 0, stream>>>(src, dst, alpha_ws, denom, node_data,
                                              out, alpha_out);
}